// Project2Dto3D_36919538876790
// MI455X (gfx1250) — compile-verified
//
#include <hip/hip_runtime.h>
#include <stdint.h>

// Problem constants (from the reference)
#define W3   240
#define H3   144
#define D3   240
#define VVOX (W3 * H3 * D3)     // 8,294,400 voxels
#define NPIX (480 * 640)        // 307,200 pixels
#define NCH  8
#define PLANE (H3 * D3)         // 34560 == H3*W3 as well (144*240)

// ---------------------------------------------------------------------------
// Kernel 1: zero-fill the 265 MB output with 128-bit stores (global_store_b128).
// Grid-stride so any grid size covers n4 exactly.
// ---------------------------------------------------------------------------
__global__ void zero_fill_f4(float4* __restrict__ out, int n4) {
    int i      = blockIdx.x * blockDim.x + threadIdx.x;
    int stride = gridDim.x * blockDim.x;
    const float4 z = make_float4(0.f, 0.f, 0.f, 0.f);
    for (; i < n4; i += stride) {
        out[i] = z;
    }
}

// ---------------------------------------------------------------------------
// Kernel 2: scatter-max. One thread per pixel, 8 channels unrolled.
//  - idx (int64) and x2d loads are fully coalesced per channel.
//  - values <= 0 can never beat the zero-initialized volume -> skip (~50%).
//  - surviving values use the native gfx1250 float-max atomic
//    (global_atomic_max_num_f32, fire-and-forget, device scope) so each
//    element is a single VMEM op, no CAS loop.
// ---------------------------------------------------------------------------
__global__ void scatter_max_proj(const float* __restrict__ x2d,
                                 const long long* __restrict__ idx,
                                 float* __restrict__ out) {
    int n = blockIdx.x * blockDim.x + threadIdx.x;
    if (n >= NPIX) return;

    // Flattened voxel index v = x*(H3*D3) + y*D3 + z, all in [0, VVOX).
    uint32_t v   = (uint32_t)idx[n];
    uint32_t x   = v / PLANE;            // mul-hi, no real divide
    uint32_t rem = v - x * PLANE;
    uint32_t y   = rem / D3;
    uint32_t z   = rem - y * D3;
    // Transposed (c, D3, H3, W3) layout: offset = z*(H3*W3) + y*W3 + x.
    uint32_t off = z * PLANE + y * W3 + x;   // H3*W3 == PLANE == 34560

#pragma unroll
    for (int c = 0; c < NCH; ++c) {
        float val = x2d[(size_t)c * NPIX + n];
        if (val > 0.0f) {
            uint64_t addr =
                (uint64_t)(uintptr_t)(out + (size_t)c * VVOX + off);
            // gfx1250 native IEEE maximumNumber atomic at device scope (L2).
            asm volatile("global_atomic_max_num_f32 %0, %1, off scope:SCOPE_DEV"
                         :
                         : "v"(addr), "v"(val)
                         : "memory");
        }
    }
}

// ---------------------------------------------------------------------------
// Launch: fill then scatter, stream-ordered.
// ---------------------------------------------------------------------------
extern "C" void kernel_launch(void* const* d_in, const int* in_sizes, int n_in,
                              void* d_out, int out_size, void* d_ws, size_t ws_size,
                              hipStream_t stream) {
    const float*     x2d = (const float*)d_in[0];      // (1,8,480,640) f32
    const long long* idx = (const long long*)d_in[1];  // (1,480,640) i64
    float*           out = (float*)d_out;              // (1,8,240,144,240) f32

    // out_size = 8 * 8,294,400 = 66,355,200 floats, divisible by 4.
    int n4 = out_size >> 2;                  // 16,588,800 float4s
    int fill_threads = 256;
    int fill_blocks  = (n4 + fill_threads * 4 - 1) / (fill_threads * 4); // ~4 stores/thread
    zero_fill_f4<<<fill_blocks, fill_threads, 0, stream>>>((float4*)out, n4);

    int sc_threads = 256;
    int sc_blocks  = (NPIX + sc_threads - 1) / sc_threads;  // 1200 blocks
    scatter_max_proj<<<sc_blocks, sc_threads, 0, stream>>>(x2d, idx, out);
}